// EigenModule_32830730010967
// MI455X (gfx1250) — compile-verified
//
#include <hip/hip_runtime.h>
#include <math.h>

typedef __attribute__((ext_vector_type(2))) float v2f;
typedef __attribute__((ext_vector_type(4))) float v4f;
typedef __attribute__((ext_vector_type(8))) float v8f;

#define WAVES_PER_BLOCK 8
#define VOX_PER_WAVE    16
#define TILES_PER_WAVE  4
#define VOX_PER_BLOCK   (WAVES_PER_BLOCK * VOX_PER_WAVE * TILES_PER_WAVE)  // 512
#define NGRAD           64
#define ROWP            68   // padded LDS row stride (floats): conflict-free A reads

__global__ __launch_bounds__(256)
void dti_eig_kernel(const float* __restrict__ dwi,
                    const float* __restrict__ Wm,     // (7,64) row-major
                    const float* __restrict__ noise,  // (nvox,3,3)
                    float* __restrict__ out,          // (nvox,3)
                    int nvox)
{
    __shared__ __align__(16) float lds[WAVES_PER_BLOCK * VOX_PER_WAVE * ROWP]; // 34816 B

    const int tid  = threadIdx.x;
    const int wv   = tid >> 5;
    const int lane = tid & 31;
    const int half = lane >> 4;   // 0: K+{0,1}, 1: K+{2,3}
    const int l16  = lane & 15;

    float* wlds = lds + wv * (VOX_PER_WAVE * ROWP);

    // ---- Branchless B-fragment preload (hoisted: reused across all tiles)
    // B[k][n] = W[n][k]; n = l16; rows >= 6 unused -> 0 (clamped addr + mask mul)
    v2f bfrag[16];
    {
        const int   nrow = (l16 < 6) ? l16 : 0;
        const float msk  = (l16 < 6) ? 1.0f : 0.0f;
        const float* wp  = Wm + nrow * NGRAD + half * 2;
        #pragma unroll
        for (int t = 0; t < 16; ++t) {
            v2f w = *(const v2f*)(wp + 4 * t);
            bfrag[t][0] = w.x * msk;
            bfrag[t][1] = w.y * msk;
        }
    }

    const int tileBase0 =
        ((int)blockIdx.x * WAVES_PER_BLOCK + wv) * (TILES_PER_WAVE * VOX_PER_WAVE);

    for (int tt = 0; tt < TILES_PER_WAVE; ++tt) {
        int tileVox = tileBase0 + tt * VOX_PER_WAVE;
        // clamp partial tail tile (workload divides exactly; safe duplicates if not)
        if (tileVox + VOX_PER_WAVE > nvox) {
            tileVox = nvox - VOX_PER_WAVE;
            if (tileVox < 0) tileVox = 0;
        }

        // ---- Phase 0: coalesced non-temporal stream of 16 vox x 64 grads; log; stage
        const v4f* dwi4 = (const v4f*)(dwi + (size_t)tileVox * NGRAD);
        #pragma unroll
        for (int i = 0; i < 8; ++i) {
            int j = i * 32 + lane;              // float4 index within tile [0,256)
            v4f v = __builtin_nontemporal_load(&dwi4[j]);
            v.x = __logf(fmaxf(v.x, 1e-9f));
            v.y = __logf(fmaxf(v.y, 1e-9f));
            v.z = __logf(fmaxf(v.z, 1e-9f));
            v.w = __logf(fmaxf(v.w, 1e-9f));
            int row = j >> 4;                   // voxel within tile
            int col = (j & 15) << 2;            // gradient index (multiple of 4)
            *(v4f*)(wlds + row * ROWP + col) = v;
        }
        __syncthreads();

        // ---- GEMV as 16 chained V_WMMA_F32_16X16X4_F32 (full fp32 precision)
        v8f acc = {0.f, 0.f, 0.f, 0.f, 0.f, 0.f, 0.f, 0.f};
        const float* arow = wlds + l16 * ROWP + half * 2;
        #pragma unroll
        for (int t = 0; t < 16; ++t) {
            v2f a;
            a[0] = arow[4 * t];
            a[1] = arow[4 * t + 1];
            acc = __builtin_amdgcn_wmma_f32_16x16x4_f32(
                /*neg_a=*/false, a, /*neg_b=*/false, bfrag[t],
                /*c_mod=*/(short)0, acc, /*reuse_a=*/false, /*reuse_b=*/false);
        }
        __syncthreads();

        // ---- Scatter fit components: lane holds n=l16, voxel rows m = half*8 + r
        float* wfit = wlds;                     // 16 voxels x 8 floats (reuse staging)
        if (l16 < 6) {
            #pragma unroll
            for (int r = 0; r < 8; ++r)
                wfit[(half * 8 + r) * 8 + l16] = acc[r];
        }
        __syncthreads();

        // ---- Per-voxel analytic 3x3 symmetric eigensolve (lanes 0..15)
        if (lane < 16) {
            int vox = tileVox + lane;
            const float* f = wfit + lane * 8;
            float f0 = f[0], f1 = f[1], f2 = f[2], f3 = f[3], f4 = f[4], f5 = f[5];

            const float* np_ = noise + (size_t)vox * 9;
            float n00 = np_[0], n01 = np_[1], n02 = np_[2];
            float n10 = np_[3], n11 = np_[4], n12 = np_[5];
            float n20 = np_[6], n21 = np_[7], n22 = np_[8];

            // D = fit[LT] + noise, then (D + D^T)/2
            float a00 = f0 + n00;
            float a11 = f2 + n11;
            float a22 = f5 + n22;
            float a01 = f1 + 0.5f * (n01 + n10);
            float a02 = f3 + 0.5f * (n02 + n20);
            float a12 = f4 + 0.5f * (n12 + n21);

            float q   = (a00 + a11 + a22) * (1.0f / 3.0f);
            float b00 = a00 - q, b11 = a11 - q, b22 = a22 - q;
            float p2  = b00 * b00 + b11 * b11 + b22 * b22
                      + 2.0f * (a01 * a01 + a02 * a02 + a12 * a12);
            float p   = sqrtf(p2 * (1.0f / 6.0f));

            float e1, e2, e3;
            if (p > 0.0f) {
                float invp = 1.0f / p;
                float c00 = b00 * invp, c11 = b11 * invp, c22 = b22 * invp;
                float c01 = a01 * invp, c02 = a02 * invp, c12 = a12 * invp;
                float halfdet = 0.5f * ( c00 * (c11 * c22 - c12 * c12)
                                       - c01 * (c01 * c22 - c12 * c02)
                                       + c02 * (c01 * c12 - c11 * c02));
                halfdet = fminf(1.0f, fmaxf(-1.0f, halfdet));
                float phi = acosf(halfdet) * (1.0f / 3.0f);
                float tp  = 2.0f * p;
                e1 = q + tp * __cosf(phi);
                e3 = q + tp * __cosf(phi + 2.0943951023931953f); // + 2*pi/3
                e2 = 3.0f * q - e1 - e3;                         // e1 >= e2 >= e3
            } else {
                e1 = e2 = e3 = q;
            }

            // reference: clip (NaN stays NaN) then where(isnan)->0 == NaN?0:max(e,1e-9)
            e1 = (e1 != e1) ? 0.0f : fmaxf(e1, 1e-9f);
            e2 = (e2 != e2) ? 0.0f : fmaxf(e2, 1e-9f);
            e3 = (e3 != e3) ? 0.0f : fmaxf(e3, 1e-9f);

            if (vox < nvox) {
                float* o = out + (size_t)vox * 3;
                o[0] = e1;
                o[1] = e2;
                o[2] = e3;
            }
        }
        __syncthreads();   // protect wfit/staging reuse before next tile's stores
    }
}

extern "C" void kernel_launch(void* const* d_in, const int* in_sizes, int n_in,
                              void* d_out, int out_size, void* d_ws, size_t ws_size,
                              hipStream_t stream) {
    const float* dwi   = (const float*)d_in[0];
    const float* Wm    = (const float*)d_in[1];
    const float* noise = (const float*)d_in[2];
    float*       out   = (float*)d_out;

    int nvox   = in_sizes[0] / NGRAD;                        // 552960
    int blocks = (nvox + VOX_PER_BLOCK - 1) / VOX_PER_BLOCK; // 1080
    dti_eig_kernel<<<dim3(blocks), dim3(256), 0, stream>>>(dwi, Wm, noise, out, nvox);
}